// GCN_50878182588434
// MI455X (gfx1250) — compile-verified
//
#include <hip/hip_runtime.h>

typedef float v2f __attribute__((ext_vector_type(2)));
typedef float v8f __attribute__((ext_vector_type(8)));

#define IN_CH 65
#define HID   64
#define KPAD  68   // K padded to multiple of 4; pad rows of W1 zeroed in LDS
#define WAVES 4
#define TPB   (WAVES * 32)

// ---------------- Phase 0: zero the aggregation workspace ----------------
__global__ void gin_zero_kernel(float* __restrict__ p, int n) {
  int i = blockIdx.x * blockDim.x + threadIdx.x;
  int stride = gridDim.x * blockDim.x;
  for (; i < n; i += stride) p[i] = 0.0f;
}

// ---------------- Phase 1: edge scatter-add (GIN aggregation) ------------
// One wave per edge; lanes cover the 65 channels (32 + 32 + 1).
__global__ void gin_scatter_kernel(const float* __restrict__ x,
                                   const long long* __restrict__ ei,
                                   float* __restrict__ agg, int nEdges) {
  const int lane = threadIdx.x & 31;
  int wid = (blockIdx.x * blockDim.x + threadIdx.x) >> 5;
  const int nwaves = (gridDim.x * blockDim.x) >> 5;
  for (int e = wid; e < nEdges; e += nwaves) {
    const long long s = ei[e];
    const long long d = ei[nEdges + e];
    const float* __restrict__ xs = x + s * IN_CH;
    float* __restrict__ ad = agg + d * IN_CH;
    atomicAdd(&ad[lane], xs[lane]);
    atomicAdd(&ad[lane + 32], xs[lane + 32]);
    if (lane == 0) atomicAdd(&ad[64], xs[64]);
  }
}

// ---------------- Phase 2: fused (x + agg) -> MLP via f32 WMMA -----------
__global__ void __launch_bounds__(TPB)
gin_mlp_kernel(const float* __restrict__ x, const float* __restrict__ agg,
               const float* __restrict__ W1, const float* __restrict__ b1,
               const float* __restrict__ W2, const float* __restrict__ b2,
               float* __restrict__ out, int nTiles, int nNodes) {
  __shared__ float W1s[KPAD * HID];          // 68x64, rows 65..67 zero
  __shared__ float W2s[HID * HID];           // 64x64
  __shared__ float stage[WAVES][16 * KPAD];  // per-wave ReLU tile, stride 68

  const int tid = threadIdx.x;

  // Cooperative load of weights into LDS (zero-pad W1 K-rows 65..67).
  for (int i = tid; i < KPAD * HID; i += TPB) {
    int k = i >> 6;                          // HID == 64
    W1s[i] = (k < IN_CH) ? W1[i] : 0.0f;     // row-major: W1[k*64+c] == W1[i]
  }
  for (int i = tid; i < HID * HID; i += TPB) W2s[i] = W2[i];
  __syncthreads();

  const int lane = tid & 31;
  const int wid  = tid >> 5;
  const int lm   = lane & 15;   // M (rows) for A, N (cols) for B/C/D
  const int lh   = lane >> 4;   // selects K pair (A/B) / +8 row offset (C/D)
  float* __restrict__ st = &stage[wid][0];

  // Per-lane bias values: lane's column within N-tile t is t*16 + lm.
  float bias1[4], bias2[4];
#pragma unroll
  for (int t = 0; t < 4; ++t) {
    bias1[t] = b1[t * 16 + lm];
    bias2[t] = b2[t * 16 + lm];
  }

  const int gw = blockIdx.x * WAVES + wid;
  const int gstride = gridDim.x * WAVES;

  for (int tile = gw; tile < nTiles; tile += gstride) {
    const int m0 = tile * 16;
    int rowA = m0 + lm;
    if (rowA >= nNodes) rowA = nNodes - 1;   // duplicate-row guard (partial tile)
    const float* __restrict__ xr = x + (long long)rowA * IN_CH;
    const float* __restrict__ ar = agg + (long long)rowA * IN_CH;

    // ---- Layer 1: D = relu((x+agg)[16x68pad] @ W1s[68x64] + b1) ----
    v8f acc[4] = {v8f{}, v8f{}, v8f{}, v8f{}};
#pragma unroll
    for (int k0 = 0; k0 < IN_CH; k0 += 4) {  // 17 steps, last one padded
      const int kA = k0 + 2 * lh;
      v2f A;
      A.x = (kA     < IN_CH) ? (xr[kA]     + ar[kA])     : 0.0f;
      A.y = (kA + 1 < IN_CH) ? (xr[kA + 1] + ar[kA + 1]) : 0.0f;
      const float* __restrict__ br = &W1s[kA * HID];
#pragma unroll
      for (int t = 0; t < 4; ++t) {
        const int col = t * 16 + lm;
        v2f B = {br[col], br[HID + col]};
        acc[t] = __builtin_amdgcn_wmma_f32_16x16x4_f32(
            false, A, false, B, (short)0, acc[t], false, false);
      }
    }

    // Epilogue: bias + ReLU, stage into LDS in row-major [16][KPAD].
#pragma unroll
    for (int t = 0; t < 4; ++t) {
      const int col = t * 16 + lm;
#pragma unroll
      for (int r = 0; r < 8; ++r) {
        const float v = acc[t][r] + bias1[t];
        st[(r + 8 * lh) * KPAD + col] = fmaxf(v, 0.0f);
      }
    }
    // Per-wave LDS ordering: drain DS ops, fence compiler reordering.
    asm volatile("s_wait_dscnt 0" ::: "memory");

    // ---- Layer 2: out = stage[16x64] @ W2s[64x64] + b2 ----
    v8f acc2[4] = {v8f{}, v8f{}, v8f{}, v8f{}};
#pragma unroll
    for (int k0 = 0; k0 < HID; k0 += 4) {    // 16 steps
      const int kA = k0 + 2 * lh;
      v2f A = {st[lm * KPAD + kA], st[lm * KPAD + kA + 1]};
#pragma unroll
      for (int t = 0; t < 4; ++t) {
        const int col = t * 16 + lm;
        v2f B = {W2s[kA * HID + col], W2s[(kA + 1) * HID + col]};
        acc2[t] = __builtin_amdgcn_wmma_f32_16x16x4_f32(
            false, A, false, B, (short)0, acc2[t], false, false);
      }
    }

#pragma unroll
    for (int t = 0; t < 4; ++t) {
      const int col = t * 16 + lm;
#pragma unroll
      for (int r = 0; r < 8; ++r) {
        const int row = m0 + r + 8 * lh;
        if (row < nNodes) out[(long long)row * HID + col] = acc2[t][r] + bias2[t];
      }
    }
    asm volatile("s_wait_dscnt 0" ::: "memory");  // stage reads done before reuse
  }
}

// ---------------- Launch ----------------
extern "C" void kernel_launch(void* const* d_in, const int* in_sizes, int n_in,
                              void* d_out, int out_size, void* d_ws, size_t ws_size,
                              hipStream_t stream) {
  (void)n_in; (void)out_size; (void)ws_size;
  const float*     x  = (const float*)d_in[0];
  const long long* ei = (const long long*)d_in[1];   // int64 edge_index [2, E]
  const float*     W1 = (const float*)d_in[2];
  const float*     b1 = (const float*)d_in[3];
  const float*     W2 = (const float*)d_in[4];
  const float*     b2 = (const float*)d_in[5];
  float* out = (float*)d_out;

  const int nNodes = in_sizes[0] / IN_CH;   // 100000
  const int nEdges = in_sizes[1] / 2;       // 1600000
  float* agg = (float*)d_ws;                // [nNodes, IN_CH] f32 scratch

  // Phase 0: zero aggregation buffer (deterministic each call).
  {
    const int n = nNodes * IN_CH;
    const int blocks = (n + 255) / 256;
    gin_zero_kernel<<<blocks, 256, 0, stream>>>(agg, n);
  }

  // Phase 1: scatter-add over edges (one wave per edge; 8 waves / 256-thr block).
  {
    const int blocks = (nEdges + 7) / 8;
    gin_scatter_kernel<<<blocks, 256, 0, stream>>>(x, ei, agg, nEdges);
  }

  // Phase 2: fused add + 2-layer MLP with f32 WMMA. One wave per 16-node tile.
  {
    const int nTiles = (nNodes + 15) / 16;  // 6250
    const int blocks = (nTiles + WAVES - 1) / WAVES;
    gin_mlp_kernel<<<blocks, TPB, 0, stream>>>(x, agg, W1, b1, W2, b2,
                                               out, nTiles, nNodes);
  }
}